// Wave_generator_21973052686533
// MI455X (gfx1250) — compile-verified
//
#include <hip/hip_runtime.h>
#include <hip/hip_bf16.h>
#include <math.h>

typedef __attribute__((ext_vector_type(2))) float v2f;
typedef __attribute__((ext_vector_type(8))) float v8f;

#define S_TOT 4096
#define DMODEL 128
#define NHEADS 4
#define NCOE 50
#define NSCALES 4

// small-scratch layout (floats), offset from ws + 7*S*D
#define OFF_COLSUM 0
#define OFF_CSC 128        // coe_scaling [50]
#define OFF_CWV 178        // coe_wavelet [50]
#define OFF_CSS 228        // coe_scales  [4]

// ---------------------------------------------------------------------------
// Kernel 1: sine encoding + GEMM  eig = eeig(129) @ W(129x128) + b
// K padded 129 -> 132; LDS row stride 134
// ---------------------------------------------------------------------------
__global__ void sine_gemm_kernel(const float* __restrict__ ev,
                                 const float* __restrict__ W,
                                 const float* __restrict__ bias,
                                 float* __restrict__ C)
{
    __shared__ float As[16 * 134];
    const int m0 = blockIdx.x * 16;
    const int tid = threadIdx.x;
    const float fac = -9.210340371976184f / 128.0f; // -ln(10000)/HIDDEN

    for (int idx = tid; idx < 16 * 132; idx += blockDim.x) {
        int r = idx / 132, c = idx - r * 132;
        float e = ev[m0 + r];
        float val;
        if (c == 0)        val = e;
        else if (c <= 64)  val = sinf(e * 100.0f * expf((float)(2 * (c - 1)) * fac));
        else if (c <= 128) val = cosf(e * 100.0f * expf((float)(2 * (c - 65)) * fac));
        else               val = 0.0f;
        As[r * 134 + c] = val;
    }
    __syncthreads();

    const int wave = tid >> 5, lane = tid & 31;
    const int half = lane >> 4, l16 = lane & 15;
    const int nwaves = blockDim.x >> 5;

    for (int nt = wave; nt < (DMODEL >> 4); nt += nwaves) {
        const int col = nt * 16 + l16;
        v8f acc = {};
        for (int k0 = 0; k0 < 132; k0 += 4) {
            int ka = k0 + half * 2;
            v2f a, b;
            a.x = As[l16 * 134 + ka];
            a.y = As[l16 * 134 + ka + 1];
            b.x = (ka     < 129) ? W[ka * DMODEL + col]       : 0.0f;
            b.y = (ka + 1 < 129) ? W[(ka + 1) * DMODEL + col] : 0.0f;
            acc = __builtin_amdgcn_wmma_f32_16x16x4_f32(false, a, false, b,
                                                        (short)0, acc, false, false);
        }
        for (int r = 0; r < 8; ++r) {
            int row = m0 + r + half * 8;
            C[row * DMODEL + col] = acc[r] + bias[col];
        }
    }
}

// ---------------------------------------------------------------------------
// Kernel 2: LayerNorm, one wave32 per row of 128
// ---------------------------------------------------------------------------
__global__ void ln_kernel(const float* __restrict__ x,
                          const float* __restrict__ g,
                          const float* __restrict__ b,
                          float* __restrict__ y)
{
    int wave = (blockIdx.x * blockDim.x + threadIdx.x) >> 5;
    int lane = threadIdx.x & 31;
    if (wave >= S_TOT) return;
    const float* row = x + wave * DMODEL;
    float v[4], s = 0.0f, sq = 0.0f;
    for (int i = 0; i < 4; ++i) {
        float t = row[lane + 32 * i];
        v[i] = t; s += t; sq += t * t;
    }
    for (int off = 16; off; off >>= 1) {
        s  += __shfl_xor(s, off, 32);
        sq += __shfl_xor(sq, off, 32);
    }
    float m   = s * (1.0f / 128.0f);
    float var = sq * (1.0f / 128.0f) - m * m;
    float inv = rsqrtf(var + 1e-5f);
    float* yr = y + wave * DMODEL;
    for (int i = 0; i < 4; ++i) {
        int c = lane + 32 * i;
        yr[c] = (v[i] - m) * inv * g[c] + b[c];
    }
}

// ---------------------------------------------------------------------------
// Kernel 3: generic WMMA GEMM  C[S x N] = A[S x K] @ W[K x N] + bias
//           optional exact GELU, optional residual add. 16 rows per block.
// K <= 132, multiple of 4.
// ---------------------------------------------------------------------------
__global__ void gemm16_kernel(const float* __restrict__ A,
                              const float* __restrict__ W,
                              const float* __restrict__ bias,
                              const float* __restrict__ R,
                              float* __restrict__ C,
                              int K, int N, int act)
{
    __shared__ float As[16 * 134];
    const int Kp = K + 2;
    const int m0 = blockIdx.x * 16;
    const int tid = threadIdx.x;

    for (int idx = tid; idx < 16 * K; idx += blockDim.x) {
        int r = idx / K, c = idx - r * K;
        As[r * Kp + c] = A[(m0 + r) * K + c];
    }
    __syncthreads();

    const int wave = tid >> 5, lane = tid & 31;
    const int half = lane >> 4, l16 = lane & 15;
    const int nwaves = blockDim.x >> 5;
    const int ntiles = N >> 4;

    for (int nt = wave; nt < ntiles; nt += nwaves) {
        const int col = nt * 16 + l16;
        v8f acc = {};
        for (int k0 = 0; k0 < K; k0 += 4) {
            int ka = k0 + half * 2;
            v2f a, b;
            a.x = As[l16 * Kp + ka];
            a.y = As[l16 * Kp + ka + 1];
            b.x = W[ka * N + col];
            b.y = W[(ka + 1) * N + col];
            acc = __builtin_amdgcn_wmma_f32_16x16x4_f32(false, a, false, b,
                                                        (short)0, acc, false, false);
        }
        for (int r = 0; r < 8; ++r) {
            int row = m0 + r + half * 8;
            float v = acc[r] + bias[col];
            if (act == 1) v = 0.5f * v * (1.0f + erff(v * 0.70710678118654752f));
            if (R) v += R[row * N + col];
            C[row * N + col] = v;
        }
    }
}

// ---------------------------------------------------------------------------
// Kernel 4: flash attention. Block = 4 waves (one per head), 16 queries/block.
// qkv row-major [S][384]; q at +0, k at +128, v at +256 (head h uses h*32..).
// ---------------------------------------------------------------------------
__global__ void attn_kernel(const float* __restrict__ qkv,
                            float* __restrict__ o,
                            const int* __restrict__ sele_num)
{
    __shared__ float Plds[NHEADS][16 * 17];
    const int m0   = blockIdx.x * 16;
    const int h    = threadIdx.x >> 5;
    const int lane = threadIdx.x & 31;
    const int half = lane >> 4, l16 = lane & 15;
    const int sele = sele_num[0];
    const int QOFF = h * 32, KOFF = DMODEL + h * 32, VOFF = 2 * DMODEL + h * 32;
    const float scale = 0.17677669529663689f; // 1/sqrt(32)

    // preload Q fragments for this lane (A layout: 16 floats)
    float qreg[16];
    {
        const float* qrow = qkv + (m0 + l16) * 384 + QOFF + half * 2;
        for (int i = 0; i < 8; ++i) {
            qreg[2 * i]     = qrow[4 * i];
            qreg[2 * i + 1] = qrow[4 * i + 1];
        }
    }

    v8f o0 = {}, o1 = {};
    float mrow[8], lrow[8];
    for (int r = 0; r < 8; ++r) { mrow[r] = -3.0e38f; lrow[r] = 0.0f; }
    float* P = &Plds[h][0];

    for (int kt = 0; kt < S_TOT / 16; ++kt) {
        const int kb = kt * 16;
        // ---- scores: Q(16x32) @ K^T(32x16) ----
        v8f sacc = {};
        const float* krow = qkv + (kb + l16) * 384 + KOFF + half * 2;
        for (int i = 0; i < 8; ++i) {
            v2f a, b;
            a.x = qreg[2 * i];     a.y = qreg[2 * i + 1];
            b.x = krow[4 * i];     b.y = krow[4 * i + 1];
            sacc = __builtin_amdgcn_wmma_f32_16x16x4_f32(false, a, false, b,
                                                         (short)0, sacc, false, false);
        }
        // ---- online softmax per row ----
        const bool masked = (kb + l16) >= sele;
        for (int r = 0; r < 8; ++r) {
            float val = masked ? -3.0e38f : sacc[r] * scale;
            float mt = val;
            mt = fmaxf(mt, __shfl_xor(mt, 1, 16));
            mt = fmaxf(mt, __shfl_xor(mt, 2, 16));
            mt = fmaxf(mt, __shfl_xor(mt, 4, 16));
            mt = fmaxf(mt, __shfl_xor(mt, 8, 16));
            float mnew  = fmaxf(mrow[r], mt);
            float alpha = expf(mrow[r] - mnew);
            float p = masked ? 0.0f : expf(val - mnew);
            float rs = p;
            rs += __shfl_xor(rs, 1, 16);
            rs += __shfl_xor(rs, 2, 16);
            rs += __shfl_xor(rs, 4, 16);
            rs += __shfl_xor(rs, 8, 16);
            lrow[r] = lrow[r] * alpha + rs;
            mrow[r] = mnew;
            o0[r] *= alpha;
            o1[r] *= alpha;
            P[(r + half * 8) * 17 + l16] = p; // C layout -> LDS (row-major 16x16)
        }
        // ---- O += P(16x16) @ V(16x32) ----
        for (int k0 = 0; k0 < 16; k0 += 4) {
            int ka = k0 + half * 2;
            v2f a, b0, b1;
            a.x = P[l16 * 17 + ka];
            a.y = P[l16 * 17 + ka + 1];
            const float* vr0 = qkv + (kb + ka)     * 384 + VOFF;
            const float* vr1 = qkv + (kb + ka + 1) * 384 + VOFF;
            b0.x = vr0[l16];      b0.y = vr1[l16];
            b1.x = vr0[16 + l16]; b1.y = vr1[16 + l16];
            o0 = __builtin_amdgcn_wmma_f32_16x16x4_f32(false, a, false, b0,
                                                       (short)0, o0, false, false);
            o1 = __builtin_amdgcn_wmma_f32_16x16x4_f32(false, a, false, b1,
                                                       (short)0, o1, false, false);
        }
    }
    for (int r = 0; r < 8; ++r) {
        int row = m0 + r + half * 8;
        float invl = (lrow[r] > 0.0f) ? 1.0f / lrow[r] : 0.0f;
        o[row * DMODEL + h * 32 + l16]      = o0[r] * invl;
        o[row * DMODEL + h * 32 + 16 + l16] = o1[r] * invl;
    }
}

// ---------------------------------------------------------------------------
// Kernel 5: pooled projections -> coe_scaling/coe_wavelet/coe_scales
// One block, 128 threads.  sum_s(xW + b) = (sum_s x)W + cnt*b
// ---------------------------------------------------------------------------
__global__ void pool_kernel(const float* __restrict__ xf,
                            const float* __restrict__ dscW, const float* __restrict__ dscb,
                            const float* __restrict__ dwvW, const float* __restrict__ dwvb,
                            const float* __restrict__ dssW, const float* __restrict__ dssb,
                            const int* __restrict__ length, const int* __restrict__ sele_num,
                            float* __restrict__ small)
{
    __shared__ float cs[DMODEL];
    __shared__ float sig_sc[NCOE], sig_wv[NCOE], sig_ss[NSCALES];
    const int t = threadIdx.x;
    int sele = sele_num[0]; if (sele > S_TOT) sele = S_TOT;
    const float cnt = (float)sele;
    const float inv = 1.0f / ((float)length[0] + 1e-8f);

    float acc = 0.0f;
    for (int s = 0; s < sele; ++s) acc += xf[s * DMODEL + t];
    cs[t] = acc;
    __syncthreads();

    if (t < NCOE) {
        float p = dscb[t] * cnt;
        for (int d = 0; d < DMODEL; ++d) p += cs[d] * dscW[d * NCOE + t];
        p *= inv;
        sig_sc[t] = 1.0f / (1.0f + expf(-p));
    } else if (t < 2 * NCOE) {
        int j = t - NCOE;
        float p = dwvb[j] * cnt;
        for (int d = 0; d < DMODEL; ++d) p += cs[d] * dwvW[d * NCOE + j];
        p *= inv;
        sig_wv[j] = 1.0f / (1.0f + expf(-p));
    } else if (t < 2 * NCOE + NSCALES) {
        int j = t - 2 * NCOE;
        float p = dssb[j] * cnt;
        for (int d = 0; d < DMODEL; ++d) p += cs[d] * dssW[d * NSCALES + j];
        p *= inv;
        sig_ss[j] = 1.0f / (1.0f + expf(-p));
    }
    __syncthreads();

    if (t < NCOE) {
        float ssum = 0.0f;
        for (int c = 0; c < NCOE; ++c) ssum += sig_sc[c];
        small[OFF_CSC + t] = sig_sc[t] / (ssum + 1e-8f);
    } else if (t < 2 * NCOE) {
        int j = t - NCOE;
        float ssum = 0.0f;
        for (int c = 0; c < NCOE; ++c) ssum += sig_wv[c];
        small[OFF_CWV + j] = sig_wv[j] / (ssum + 1e-8f);
    } else if (t < 2 * NCOE + NSCALES) {
        small[OFF_CSS + (t - 2 * NCOE)] = sig_ss[t - 2 * NCOE] * 5.0f; // THRE
    }
}

// ---------------------------------------------------------------------------
// Kernel 6: Chebyshev bases + weighted sums + L2 normalize -> out [S x 5]
// ---------------------------------------------------------------------------
__global__ void final_kernel(const float* __restrict__ ev,
                             const float* __restrict__ small,
                             float* __restrict__ out)
{
    __shared__ float csc[NCOE], cwv[NCOE], css[NSCALES];
    const int t = threadIdx.x;
    if (t < NCOE)                  csc[t]            = small[OFF_CSC + t];
    else if (t < 2 * NCOE)         cwv[t - NCOE]     = small[OFF_CWV + t - NCOE];
    else if (t < 2 * NCOE + NSCALES) css[t - 2*NCOE] = small[OFF_CSS + t - 2 * NCOE];
    __syncthreads();

    const int s = blockIdx.x * blockDim.x + t;
    if (s >= S_TOT) return;
    const float e = ev[s];

    // scaling branch: T1, T3, ..., T99 of (e-1)
    float y = e - 1.0f;
    float te = 1.0f, to = y;
    float accs = csc[0] * 0.5f * (1.0f - to);
    for (int c = 1; c < NCOE; ++c) {
        te = 2.0f * y * to - te;
        to = 2.0f * y * te - to;
        accs += csc[c] * 0.5f * (1.0f - to);
    }
    // wavelet branch: T0, T2, ..., T98 of (fsw-1), fsw zeroed where > 2
    float aw[NSCALES];
    for (int j = 0; j < NSCALES; ++j) {
        float fs = e * css[j];
        fs = (fs > 2.0f) ? 0.0f : fs;
        float yw = fs - 1.0f;
        float te2 = 1.0f, to2 = yw;
        float acc = cwv[0] * 0.5f * (1.0f - te2);
        for (int c = 1; c < NCOE; ++c) {
            te2 = 2.0f * yw * to2 - te2;
            to2 = 2.0f * yw * te2 - to2;
            acc += cwv[c] * 0.5f * (1.0f - te2);
        }
        aw[j] = acc;
    }
    float nrm = accs * accs;
    for (int j = 0; j < NSCALES; ++j) nrm += aw[j] * aw[j];
    float invn = 1.0f / (sqrtf(nrm) + 1e-8f);
    out[s * 5 + 0] = accs * invn;
    for (int j = 0; j < NSCALES; ++j) out[s * 5 + 1 + j] = aw[j] * invn;
}

// ---------------------------------------------------------------------------
extern "C" void kernel_launch(void* const* d_in, const int* in_sizes, int n_in,
                              void* d_out, int out_size, void* d_ws, size_t ws_size,
                              hipStream_t stream)
{
    const float* eigenvalue = (const float*)d_in[0];
    const int*   length     = (const int*)  d_in[1];
    const int*   sele_num   = (const int*)  d_in[2];
    const float* eig_w_W    = (const float*)d_in[3];
    const float* eig_w_b    = (const float*)d_in[4];
    const float* mha_ln_g   = (const float*)d_in[5];
    const float* mha_ln_b   = (const float*)d_in[6];
    const float* in_W       = (const float*)d_in[7];
    const float* in_b       = (const float*)d_in[8];
    const float* out_W      = (const float*)d_in[9];
    const float* out_b      = (const float*)d_in[10];
    const float* ffn_ln_g   = (const float*)d_in[11];
    const float* ffn_ln_b   = (const float*)d_in[12];
    const float* ffn1_W     = (const float*)d_in[13];
    const float* ffn1_b     = (const float*)d_in[14];
    const float* ffn2_W     = (const float*)d_in[15];
    const float* ffn2_b     = (const float*)d_in[16];
    const float* dsc_W      = (const float*)d_in[17];
    const float* dsc_b      = (const float*)d_in[18];
    const float* dwv_W      = (const float*)d_in[19];
    const float* dwv_b      = (const float*)d_in[20];
    const float* dss_W      = (const float*)d_in[21];
    const float* dss_b      = (const float*)d_in[22];

    float* out = (float*)d_out;
    float* ws  = (float*)d_ws;

    const int SD = S_TOT * DMODEL;
    float* eig        = ws;            // SD
    float* ln         = ws + SD;       // SD (reused for ln1 and ln2)
    float* qkv        = ws + 2 * SD;   // 3*SD
    float* o          = ws + 5 * SD;   // SD
    float* eig2       = ws + 6 * SD;   // SD
    float* f1         = qkv;           // reuse: qkv dead after attention/out-proj
    float* eig_filter = o;             // reuse: o dead after out-proj
    float* small      = ws + 7 * SD;   // colsum + coes

    const int MB = S_TOT / 16; // 256 row-tile blocks

    // 1) eig = sine_encoding(e) @ eig_w_W + b
    sine_gemm_kernel<<<MB, 256, 0, stream>>>(eigenvalue, eig_w_W, eig_w_b, eig);
    // 2) ln1 = LN(eig)
    ln_kernel<<<S_TOT / 4, 128, 0, stream>>>(eig, mha_ln_g, mha_ln_b, ln);
    // 3) qkv = ln1 @ in_W + in_b
    gemm16_kernel<<<MB, 256, 0, stream>>>(ln, in_W, in_b, nullptr, qkv,
                                          DMODEL, 3 * DMODEL, 0);
    // 4) o = attention(qkv)
    attn_kernel<<<MB, NHEADS * 32, 0, stream>>>(qkv, o, sele_num);
    // 5) eig2 = o @ out_W + out_b + eig
    gemm16_kernel<<<MB, 256, 0, stream>>>(o, out_W, out_b, eig, eig2,
                                          DMODEL, DMODEL, 0);
    // 6) ln2 = LN(eig2)
    ln_kernel<<<S_TOT / 4, 128, 0, stream>>>(eig2, ffn_ln_g, ffn_ln_b, ln);
    // 7) f1 = gelu(ln2 @ ffn1_W + b)
    gemm16_kernel<<<MB, 256, 0, stream>>>(ln, ffn1_W, ffn1_b, nullptr, f1,
                                          DMODEL, DMODEL, 1);
    // 8) eig_filter = f1 @ ffn2_W + b + eig2
    gemm16_kernel<<<MB, 256, 0, stream>>>(f1, ffn2_W, ffn2_b, eig2, eig_filter,
                                          DMODEL, DMODEL, 0);
    // 9) pooled coefficients
    pool_kernel<<<1, 128, 0, stream>>>(eig_filter, dsc_W, dsc_b, dwv_W, dwv_b,
                                       dss_W, dss_b, length, sele_num, small);
    // 10) Chebyshev + normalize -> out [S x 5]
    final_kernel<<<S_TOT / 256, 256, 0, stream>>>(eigenvalue, small, out);
}